// TextureToImage_59846074302581
// MI455X (gfx1250) — compile-verified
//
#include <hip/hip_runtime.h>
#include <stdint.h>

// ---------------------------------------------------------------------------
// TextureToImage: batched sparse (gather * val -> scatter-add) per sample.
//   out[b, m%3, m/3] += vals[b,k] * x[b, col%3, col/3]   (m = rows[b,k])
// Memory-bound (~150MB moved, ~6.3 MFLOP): optimize pure data movement.
// gfx1250 paths used: async global->LDS double-buffered index streaming,
// s_wait_asynccnt pipelining, wave32 blocks, no-return f32 global atomics.
// ---------------------------------------------------------------------------

namespace {
constexpr int  kB   = 32;
constexpr int  kC   = 3;
constexpr int  kHW  = 512 * 512;          // 262144
constexpr int  kCHW = kC * kHW;           // 786432  (flat texture length N)
constexpr int  kOHW = 128 * 64;           // 8192
constexpr int  kM   = kC * kOHW;          // 24576   (rows per sample)
constexpr int  kNNZ = 98304;
constexpr long kTotalNNZ = (long)kB * kNNZ;      // 3145728
constexpr int  kOutElems = kB * kM;              // 786432 (result elems)

constexpr int kTile  = 1024;                      // nnz entries per tile
constexpr int kTPB   = 256;                       // 8 wave32 per block
constexpr int kEPT   = 4;                         // entries per thread (B128)
constexpr int kTilesPerBlock  = 4;
constexpr int kNumTiles       = (int)(kTotalNNZ / kTile);       // 3072
constexpr int kScatterBlocks  = kNumTiles / kTilesPerBlock;     // 768
constexpr int kTilesPerSample = kNNZ / kTile;                   // 96 (exact)
} // namespace

// Raw clang vector types (HIP_vector_type structs are rejected by
// __builtin_nontemporal_load and by the async-LDS builtin signature).
typedef int   tti_v4i __attribute__((vector_size(16)));
typedef float tti_v4f __attribute__((vector_size(16)));

// --------------------------- gfx1250 feature probes ------------------------
#if defined(__gfx1250__) && __has_builtin(__builtin_amdgcn_global_load_async_to_lds_b128)
#define TTI_HAS_ASYNC 1
#else
#define TTI_HAS_ASYNC 0
#endif

#if __has_builtin(__builtin_amdgcn_s_wait_asynccnt)
#define TTI_WAIT_ASYNC(n) __builtin_amdgcn_s_wait_asynccnt(n)
#elif defined(__gfx1250__)
#define TTI_WAIT_ASYNC(n) asm volatile("s_wait_asynccnt %0" ::"n"(n) : "memory")
#else
#define TTI_WAIT_ASYNC(n)
#endif

#define TTI_GPTR(p) ((__attribute__((address_space(1))) tti_v4i*)(p))
#define TTI_LPTR(p) ((__attribute__((address_space(3))) tti_v4i*)(p))

// ----------------- kernel 1: zero result region + copy mask -----------------
__global__ __launch_bounds__(256) void tti_init_kernel(
    const float4* __restrict__ mask, float4* __restrict__ out) {
  const int i = blockIdx.x * blockDim.x + threadIdx.x;   // [0, kOutElems/4)
  out[i] = make_float4(0.f, 0.f, 0.f, 0.f);
  out[kOutElems / 4 + i] = mask[i];
}

// ------------- kernel 2: gather/scale/scatter-add (memory bound) ------------
__global__ __launch_bounds__(kTPB) void tti_scatter_kernel(
    const float* __restrict__ x, const int* __restrict__ rows,
    const int* __restrict__ cols, const float* __restrict__ vals,
    float* __restrict__ out) {
  const int  t         = threadIdx.x;
  const long firstTile = (long)blockIdx.x * kTilesPerBlock;

#if TTI_HAS_ASYNC
  // Per-thread-owned LDS slots (no cross-wave sharing -> no barriers needed).
  __shared__ __align__(16) int   sRows[2][kTile];
  __shared__ __align__(16) int   sCols[2][kTile];
  __shared__ __align__(16) float sVals[2][kTile];

  auto stage = [&](long tile, int buf) {
    const long base = tile * kTile + (long)kEPT * t;     // 16B aligned
    __builtin_amdgcn_global_load_async_to_lds_b128(
        TTI_GPTR(rows + base), TTI_LPTR(&sRows[buf][kEPT * t]), 0, 0);
    __builtin_amdgcn_global_load_async_to_lds_b128(
        TTI_GPTR(cols + base), TTI_LPTR(&sCols[buf][kEPT * t]), 0, 0);
    __builtin_amdgcn_global_load_async_to_lds_b128(
        TTI_GPTR(vals + base), TTI_LPTR(&sVals[buf][kEPT * t]), 0, 0);
  };

  stage(firstTile, 0);                                   // prime the pipeline
  for (int i = 0; i < kTilesPerBlock; ++i) {
    const long tile = firstTile + i;
    if (i + 1 < kTilesPerBlock) {
      stage(tile + 1, (i + 1) & 1);  // issue next tile's 3 async loads first
      TTI_WAIT_ASYNC(3);             // in-order ASYNCcnt: current tile done
    } else {
      TTI_WAIT_ASYNC(0);
    }
    asm volatile("" ::: "memory");

    const int    b   = (int)(tile / kTilesPerSample);    // tile never spans b
    const float* xb  = x + (long)b * kCHW;
    float*       ob  = out + (long)b * kM;
    const int    buf = i & 1;
#pragma unroll
    for (int e = 0; e < kEPT; ++e) {
      const int      idx = kEPT * t + e;
      const unsigned um  = (unsigned)sRows[buf][idx];
      const unsigned uc  = (unsigned)sCols[buf][idx];
      const float    v   = sVals[buf][idx];
      // gather from NCHW x: col (channels-last) -> (c, pix)
      const unsigned pix = uc / 3u;
      const unsigned ch  = uc - pix * 3u;
      const float    xv  = xb[ch * (unsigned)kHW + pix];
      // scatter to NCHW out: m (channels-last) -> (c, opix)
      const unsigned opix = um / 3u;
      const unsigned oc   = um - opix * 3u;
      atomicAdd(ob + oc * (unsigned)kOHW + opix, v * xv);
    }
  }
#else
  // Fallback: direct coalesced 16B loads, non-temporal (streams read once).
  for (int i = 0; i < kTilesPerBlock; ++i) {
    const long    tile = firstTile + i;
    const long    base = tile * kTile + (long)kEPT * t;
    const tti_v4i r4 = __builtin_nontemporal_load((const tti_v4i*)(rows + base));
    const tti_v4i c4 = __builtin_nontemporal_load((const tti_v4i*)(cols + base));
    const tti_v4f v4 = __builtin_nontemporal_load((const tti_v4f*)(vals + base));
    const int     b  = (int)(tile / kTilesPerSample);
    const float*  xb = x + (long)b * kCHW;
    float*        ob = out + (long)b * kM;
#pragma unroll
    for (int e = 0; e < 4; ++e) {
      const unsigned um  = (unsigned)r4[e];
      const unsigned uc  = (unsigned)c4[e];
      const float    v   = v4[e];
      const unsigned pix = uc / 3u;
      const unsigned ch  = uc - pix * 3u;
      const float    xv  = xb[ch * (unsigned)kHW + pix];
      const unsigned opix = um / 3u;
      const unsigned oc   = um - opix * 3u;
      atomicAdd(ob + oc * (unsigned)kOHW + opix, v * xv);
    }
  }
#endif
}

// --------------------------------- launch ----------------------------------
extern "C" void kernel_launch(void* const* d_in, const int* in_sizes, int n_in,
                              void* d_out, int out_size, void* d_ws,
                              size_t ws_size, hipStream_t stream) {
  (void)in_sizes; (void)n_in; (void)out_size; (void)d_ws; (void)ws_size;
  const float* x    = (const float*)d_in[0];
  const int*   rows = (const int*)d_in[1];
  const int*   cols = (const int*)d_in[2];
  const float* vals = (const float*)d_in[3];
  const float* mask = (const float*)d_in[4];
  float*       out  = (float*)d_out;

  // result := 0, out[786432:] := mask  (786432/4 = 196608 float4s, 768 blocks)
  tti_init_kernel<<<(kOutElems / 4) / 256, 256, 0, stream>>>(
      (const float4*)mask, (float4*)out);
  // gather * val -> atomic scatter-add, async-LDS double buffered
  tti_scatter_kernel<<<kScatterBlocks, kTPB, 0, stream>>>(x, rows, cols, vals,
                                                          out);
}